// GetDensity_14233521619184
// MI455X (gfx1250) — compile-verified
//
#include <hip/hip_runtime.h>
#include <math.h>

typedef __attribute__((ext_vector_type(2))) float v2f;
typedef __attribute__((ext_vector_type(8))) float v8f;

#define K_NANG  13
#define K_NWAVE 8
#define K_NORB  32
#define K_ROW   (K_NANG * K_NWAVE)   /* 104 floats per atom row */
#define K_HYPN  (3 * K_NWAVE * K_NORB) /* 768 floats of hyper */

// ---------------------------------------------------------------- zero scratch
__global__ __launch_bounds__(256) void zero_ws_kernel(float* __restrict__ p, int n) {
    int i = blockIdx.x * blockDim.x + threadIdx.x;
    int stride = gridDim.x * blockDim.x;
    for (; i < n; i += stride) p[i] = 0.0f;
}

// ------------------------------------------------- phase 1: neighbor scatter
// One thread per neighbor pair. 104 hardware f32 atomics into L2-resident
// sum_worbital[natom][13][8].
__global__ __launch_bounds__(256) void scatter_orbitals_kernel(
    const float* __restrict__ cart,   const float* __restrict__ shifts,
    const float* __restrict__ rs,     const float* __restrict__ inta,
    const float* __restrict__ params, const int*   __restrict__ neigh,
    const int*   __restrict__ species, float* __restrict__ sw, int nneigh)
{
    const float PI_OVER_CUT = 0.62831853071795864769f; // pi / 5.0
    int n = blockIdx.x * blockDim.x + threadIdx.x;
    int stride = gridDim.x * blockDim.x;
    for (; n < nneigh; n += stride) {
        const int i = neigh[n];
        const int j = neigh[nneigh + n];
        const float vx = cart[3 * i + 0] - cart[3 * j + 0] - shifts[3 * n + 0];
        const float vy = cart[3 * i + 1] - cart[3 * j + 1] - shifts[3 * n + 1];
        const float vz = cart[3 * i + 2] - cart[3 * j + 2] - shifts[3 * n + 2];
        const float d  = sqrtf(vx * vx + vy * vy + vz * vz);
        float fc = 0.5f * __cosf(d * PI_OVER_CUT) + 0.5f;
        fc = fc * fc;

        const int sp = species[j];
        float rad[K_NWAVE];
#pragma unroll
        for (int w = 0; w < K_NWAVE; ++w) {
            const float dr = d - rs[sp * K_NWAVE + w];
            rad[w] = __expf(inta[sp * K_NWAVE + w] * dr * dr) * params[sp * K_NWAVE + w];
        }

        float ang[K_NANG];
        ang[0]  = fc;
        ang[1]  = fc * vx;  ang[2]  = fc * vy;  ang[3]  = fc * vz;
        ang[4]  = ang[1] * vx; ang[5]  = ang[1] * vy; ang[6]  = ang[1] * vz;
        ang[7]  = ang[2] * vx; ang[8]  = ang[2] * vy; ang[9]  = ang[2] * vz;
        ang[10] = ang[3] * vx; ang[11] = ang[3] * vy; ang[12] = ang[3] * vz;

        float* base = sw + (size_t)i * K_ROW;
#pragma unroll
        for (int a = 0; a < K_NANG; ++a) {
#pragma unroll
            for (int w = 0; w < K_NWAVE; ++w) {
                unsafeAtomicAdd(base + a * K_NWAVE + w, ang[a] * rad[w]);
            }
        }
    }
}

// -------------------------------------------- phase 2: per-atom WMMA density
// One wave per atom. A'[16,24] block-structured (INDEX_PARA folded in),
// B'[24,32] = vstack(hyper). 6 K-steps x 2 N-tiles of V_WMMA_F32_16X16X4_F32.
#define KSTEP(KK, P, S0, S1, S2, S3)                                           \
    {                                                                          \
        const bool match = (ip == (P));                                        \
        v2f a;                                                                 \
        a.x = match ? (hib ? (S2) : (S0)) : 0.0f;                              \
        a.y = match ? (hib ? (S3) : (S1)) : 0.0f;                              \
        const int k0 = 4 * (KK) + 2 * hi;                                      \
        v2f b0, b1;                                                            \
        b0.x = sh_h[k0 * K_NORB + m];                                          \
        b0.y = sh_h[(k0 + 1) * K_NORB + m];                                    \
        b1.x = sh_h[k0 * K_NORB + 16 + m];                                     \
        b1.y = sh_h[(k0 + 1) * K_NORB + 16 + m];                               \
        c0 = __builtin_amdgcn_wmma_f32_16x16x4_f32(false, a, false, b0,        \
                                                   (short)0, c0, false, false);\
        c1 = __builtin_amdgcn_wmma_f32_16x16x4_f32(false, a, false, b1,        \
                                                   (short)0, c1, false, false);\
    }

__global__ __launch_bounds__(256) void density_wmma_kernel(
    const float* __restrict__ sw, const float* __restrict__ hyper,
    float* __restrict__ out, int natom)
{
    __shared__ float sh_h[K_HYPN];
    for (int t = threadIdx.x; t < K_HYPN; t += blockDim.x) sh_h[t] = hyper[t];
    __syncthreads();

    const int lane = threadIdx.x & 31;
    const int wid  = threadIdx.x >> 5;
    const int atom = blockIdx.x * (blockDim.x >> 5) + wid;
    if (atom >= natom) return; // wave-uniform: EXEC stays all-ones for WMMA

    const int  m   = lane & 15;
    const int  hi  = lane >> 4;
    const bool hib = (hi != 0);

    float4 e0 = make_float4(0.f, 0.f, 0.f, 0.f);
    float4 e1 = make_float4(0.f, 0.f, 0.f, 0.f);
    if (m < K_NANG) {
        const float4* row = (const float4*)(sw + (size_t)atom * K_ROW + m * K_NWAVE);
        e0 = row[0];
        e1 = row[1];
    }
    const int ip = (m == 0) ? 0 : ((m < 4) ? 1 : 2); // INDEX_PARA[m]

    v8f c0 = {};
    v8f c1 = {};
    KSTEP(0, 0, e0.x, e0.y, e0.z, e0.w)
    KSTEP(1, 0, e1.x, e1.y, e1.z, e1.w)
    KSTEP(2, 1, e0.x, e0.y, e0.z, e0.w)
    KSTEP(3, 1, e1.x, e1.y, e1.z, e1.w)
    KSTEP(4, 2, e0.x, e0.y, e0.z, e0.w)
    KSTEP(5, 2, e1.x, e1.y, e1.z, e1.w)

    float s0 = 0.0f, s1 = 0.0f;
#pragma unroll
    for (int r = 0; r < 8; ++r) {
        s0 += c0[r] * c0[r];
        s1 += c1[r] * c1[r];
    }
    // combine M=0..7 (lanes 0-15) with M=8..15 (lanes 16-31); pad rows are 0
    s0 += __shfl_xor(s0, 16, 32);
    s1 += __shfl_xor(s1, 16, 32);

    if (lane < 16) {
        out[(size_t)atom * K_NORB + m]      = s0;
        out[(size_t)atom * K_NORB + 16 + m] = s1;
    }
}

// ----------------------------------------------------------------- launcher
extern "C" void kernel_launch(void* const* d_in, const int* in_sizes, int n_in,
                              void* d_out, int out_size, void* d_ws, size_t ws_size,
                              hipStream_t stream) {
    const float* cart    = (const float*)d_in[0];
    const float* shifts  = (const float*)d_in[1];
    const float* rs      = (const float*)d_in[2];
    const float* inta    = (const float*)d_in[3];
    const float* params  = (const float*)d_in[4];
    const float* hyper   = (const float*)d_in[5];
    const int*   neigh   = (const int*)d_in[6];
    const int*   species = (const int*)d_in[7];
    float*       out     = (float*)d_out;

    const int natom  = in_sizes[0] / 3;
    const int nneigh = in_sizes[6] / 2;

    float* sw = (float*)d_ws;                 // sum_worbital[natom][13][8]
    const int swn = natom * K_ROW;

    {
        int blocks = (swn + 255) / 256;
        zero_ws_kernel<<<blocks, 256, 0, stream>>>(sw, swn);
    }
    {
        int blocks = (nneigh + 255) / 256;
        scatter_orbitals_kernel<<<blocks, 256, 0, stream>>>(
            cart, shifts, rs, inta, params, neigh, species, sw, nneigh);
    }
    {
        const int wavesPerBlock = 256 / 32;
        int blocks = (natom + wavesPerBlock - 1) / wavesPerBlock;
        density_wmma_kernel<<<blocks, 256, 0, stream>>>(sw, hyper, out, natom);
    }
}